// GCN_9929964388353
// MI455X (gfx1250) — compile-verified
//
#include <hip/hip_runtime.h>
#include <math.h>

typedef __attribute__((ext_vector_type(2))) float v2f;
typedef __attribute__((ext_vector_type(4))) float v4f;
typedef __attribute__((ext_vector_type(8))) float v8f;

#define IN_DIM 128
#define HID    64
#define RS     80   // LDS row stride in float2 units: 640B = 128 mod 256 -> no bank conflicts

// ---------------- utility ----------------
__global__ void k_zero(float* __restrict__ p, int n) {
  int i = blockIdx.x * blockDim.x + threadIdx.x;
  int stride = gridDim.x * blockDim.x;
  for (; i < n; i += stride) p[i] = 0.0f;
}

// deg[v] += 1 for each incoming edge (self-loop added analytically later)
__global__ void k_deg(const long long* __restrict__ dst, int E, float* __restrict__ deg) {
  int e = blockIdx.x * blockDim.x + threadIdx.x;
  if (e < E) unsafeAtomicAdd(&deg[(int)dst[e]], 1.0f);
}

// dinv[v] = rsqrt(deg[v] + 1)   (self-loop guarantees deg_total >= 1)
__global__ void k_dinv(float* __restrict__ degdinv, int N) {
  int v = blockIdx.x * blockDim.x + threadIdx.x;
  if (v < N) degdinv[v] = rsqrtf(degdinv[v] + 1.0f);
}

// ---------------- layer 1 GEMM: hlin = x @ W1 (fp32 WMMA) ----------------
// One wave computes a 16x64 output tile with 4 16x16 accumulators.
// A (16x4 f32): lane<16 holds M=lane,(K0,K1); lane>=16 holds M=lane-16,(K2,K3).
// B (4x16 f32): mirrored — lane<16 holds N=lane rows (K0,K1); lane>=16 rows (K2,K3).
// W1 staged in LDS pair-interleaved: sW2[kpair*RS + col] = {W[2kp][col], W[2kp+1][col]}
// so each lane's B operand is a single aligned ds_load_b64 into an even VGPR pair.
// C/D (16x16 f32): VGPR r: lanes0-15 -> M=r, lanes16-31 -> M=r+8, N=lane&15.
__global__ void __launch_bounds__(256) k_gemm1_wmma(const float* __restrict__ x,
                                                    const float* __restrict__ W1,
                                                    float* __restrict__ hlin, int N) {
  __shared__ float sW[(IN_DIM / 2) * RS * 2];  // 64 kpairs * 80 float2 = 40 KB
  {
    v2f* s2 = (v2f*)sW;
    for (int i = threadIdx.x; i < (IN_DIM / 2) * HID; i += 256) {
      int kp  = i >> 6;       // 0..63
      int col = i & 63;       // 0..63
      v2f p = { W1[(2 * kp) * HID + col], W1[(2 * kp + 1) * HID + col] };
      s2[kp * RS + col] = p;
    }
  }
  __syncthreads();

  const int wave = threadIdx.x >> 5;
  const int lane = threadIdx.x & 31;
  const int m0 = (blockIdx.x * 8 + wave) * 16;
  if (m0 >= N) return;  // wave-uniform: EXEC stays all-1s for WMMA

  const int half  = lane >> 4;   // 0 or 1
  const int l16   = lane & 15;
  const int khalf = half * 2;    // this half-wave carries K+2,K+3
  const float* arow = x + (size_t)(m0 + l16) * IN_DIM;
  const v2f* s2 = (const v2f*)sW;

  v8f acc0 = {}, acc1 = {}, acc2 = {}, acc3 = {};
  for (int k0 = 0; k0 < IN_DIM; k0 += 4) {
    v2f a = *(const v2f*)(arow + k0 + khalf);
    const v2f* brow = s2 + ((k0 >> 1) + half) * RS + l16;
    v2f b0 = brow[0];
    v2f b1 = brow[16];
    v2f b2 = brow[32];
    v2f b3 = brow[48];
    acc0 = __builtin_amdgcn_wmma_f32_16x16x4_f32(false, a, false, b0, (short)0, acc0, false, false);
    acc1 = __builtin_amdgcn_wmma_f32_16x16x4_f32(false, a, false, b1, (short)0, acc1, false, false);
    acc2 = __builtin_amdgcn_wmma_f32_16x16x4_f32(false, a, false, b2, (short)0, acc2, false, false);
    acc3 = __builtin_amdgcn_wmma_f32_16x16x4_f32(false, a, false, b3, (short)0, acc3, false, false);
  }

  for (int r = 0; r < 8; ++r) {
    float* orow = hlin + (size_t)(m0 + r + half * 8) * HID + l16;
    orow[0]  = acc0[r];
    orow[16] = acc1[r];
    orow[32] = acc2[r];
    orow[48] = acc3[r];
  }
}

// ---------------- layer 1 aggregation (dominant kernel) ----------------
// 64 threads per edge: coalesced 256B gather of hlin[src], hw f32 atomic scatter.
__global__ void k_agg1(const long long* __restrict__ src, const long long* __restrict__ dst,
                       int E, const float* __restrict__ hlin, const float* __restrict__ dinv,
                       float* __restrict__ agg) {
  int g = blockIdx.x * blockDim.x + threadIdx.x;
  int e = g >> 6;
  int i = g & 63;
  if (e >= E) return;
  int s = (int)src[e], d = (int)dst[e];
  float norm = dinv[s] * dinv[d];
  float val = hlin[(size_t)s * HID + i] * norm;
  unsafeAtomicAdd(&agg[(size_t)d * HID + i], val);
}

// h = relu(agg + self_loop_term + b1), written in place over agg
__global__ void k_fin1(float* __restrict__ agg, const float* __restrict__ hlin,
                       const float* __restrict__ dinv, const float* __restrict__ b1, int N) {
  int idx = blockIdx.x * blockDim.x + threadIdx.x;
  if (idx >= N * HID) return;
  int v = idx >> 6;
  int i = idx & 63;
  float dv = dinv[v];
  float val = agg[idx] + hlin[idx] * dv * dv + b1[i];
  agg[idx] = fmaxf(val, 0.0f);
}

// ---------------- layer 2: hdot = h @ W2 (64-wide dot per node) ----------------
__global__ void k_dot2(const float* __restrict__ h, const float* __restrict__ W2,
                       float* __restrict__ hdot, int N) {
  int wave = threadIdx.x >> 5;
  int lane = threadIdx.x & 31;
  int v = blockIdx.x * 8 + wave;
  if (v >= N) return;
  v2f hv = *(const v2f*)(h + (size_t)v * HID + lane * 2);
  v2f w  = *(const v2f*)(W2 + lane * 2);
  float s = hv.x * w.x + hv.y * w.y;
  for (int off = 16; off > 0; off >>= 1) s += __shfl_xor(s, off, 32);
  if (lane == 0) hdot[v] = s;
}

__global__ void k_agg2(const long long* __restrict__ src, const long long* __restrict__ dst,
                       int E, const float* __restrict__ hdot, const float* __restrict__ dinv,
                       float* __restrict__ agg2) {
  int e = blockIdx.x * blockDim.x + threadIdx.x;
  if (e >= E) return;
  int s = (int)src[e], d = (int)dst[e];
  unsafeAtomicAdd(&agg2[d], hdot[s] * dinv[s] * dinv[d]);
}

__global__ void k_fin2(const float* __restrict__ agg2, const float* __restrict__ hdot,
                       const float* __restrict__ dinv, const float* __restrict__ b2,
                       float* __restrict__ out, int N) {
  int v = blockIdx.x * blockDim.x + threadIdx.x;
  if (v >= N) return;
  float dv = dinv[v];
  float val = agg2[v] + hdot[v] * dv * dv + b2[0];
  out[v] = 1.0f / (1.0f + expf(-val));
}

// ---------------- launch ----------------
extern "C" void kernel_launch(void* const* d_in, const int* in_sizes, int n_in,
                              void* d_out, int out_size, void* d_ws, size_t ws_size,
                              hipStream_t stream) {
  const float*     x   = (const float*)d_in[0];
  const long long* ei  = (const long long*)d_in[1];  // int64 [2, E]
  const float*     W1  = (const float*)d_in[2];
  const float*     b1  = (const float*)d_in[3];
  const float*     W2  = (const float*)d_in[4];
  const float*     b2  = (const float*)d_in[5];

  const int N = in_sizes[0] / IN_DIM;   // 100000
  const int E = in_sizes[1] / 2;        // 1600000
  const long long* src = ei;
  const long long* dst = ei + E;

  // workspace layout (floats): dinv[N] | hlin[N*HID] | agg1[N*HID]
  // after layer 1: hlin region reused as hdot[N] | agg2[N]
  float* ws   = (float*)d_ws;
  float* dinv = ws;
  float* hlin = ws + N;
  float* agg1 = hlin + (size_t)N * HID;
  float* hdot = hlin;
  float* agg2 = hlin + N;

  k_zero<<<512, 256, 0, stream>>>(dinv, N);
  k_zero<<<2048, 256, 0, stream>>>(agg1, N * HID);
  k_deg<<<(E + 255) / 256, 256, 0, stream>>>(dst, E, dinv);
  k_dinv<<<(N + 255) / 256, 256, 0, stream>>>(dinv, N);

  k_gemm1_wmma<<<(N / 16 + 7) / 8, 256, 0, stream>>>(x, W1, hlin, N);

  int agg1_threads = E * 64;
  k_agg1<<<(agg1_threads + 255) / 256, 256, 0, stream>>>(src, dst, E, hlin, dinv, agg1);
  k_fin1<<<(N * HID + 255) / 256, 256, 0, stream>>>(agg1, hlin, dinv, b1, N);

  k_zero<<<512, 256, 0, stream>>>(agg2, N);
  k_dot2<<<(N + 7) / 8, 256, 0, stream>>>(agg1, W2, hdot, N);
  k_agg2<<<(E + 255) / 256, 256, 0, stream>>>(src, dst, E, hdot, dinv, agg2);
  k_fin2<<<(N + 255) / 256, 256, 0, stream>>>(agg2, hdot, dinv, b2, (float*)d_out, N);
}